// iDCT_Channel_59571196395681
// MI455X (gfx1250) — compile-verified
//
#include <hip/hip_runtime.h>
#include <hip/hip_bf16.h>
#include <math.h>

// ---------------------------------------------------------------------------
// iDCT along channel axis == DCT-III matrix multiply:
//   out[c] = x[0] + 2 * sum_{k>=1} x[k] * cos(pi*k*(c+0.5)/C)
// GEMM: M[256x256] (A) x X[256 x (b*H*W)] (B), K = 256, N = 131072.
// fp32 WMMA (V_WMMA_F32_16X16X4_F32) keeps full precision; op is memory-bound
// (AI ~16 FLOP/B, HBM floor ~46us at 23.3 TB/s).
// B tiles staged to LDS via GLOBAL_LOAD_ASYNC_TO_LDS_B128 (ASYNCcnt path).
// ---------------------------------------------------------------------------

#define C_DIM 256
#define HW    16384          // 128*128
#define KC    32             // K chunk staged per LDS buffer
#define NT    32             // pixels per block
#define LDSS  40             // padded LDS row stride (floats): 2*40 mod 64 = 16 -> disjoint bank runs

typedef float v2f __attribute__((ext_vector_type(2)));
typedef float v8f __attribute__((ext_vector_type(8)));
typedef int   v4i __attribute__((ext_vector_type(4)));

typedef __attribute__((address_space(1))) v4i* gv4i_p;   // global int4*
typedef __attribute__((address_space(3))) v4i* lv4i_p;   // LDS int4*

#if defined(__has_builtin)
#  if __has_builtin(__builtin_amdgcn_global_load_async_to_lds_b128)
#    define HAVE_ASYNC_BUILTIN 1
#  endif
#  if __has_builtin(__builtin_amdgcn_s_wait_asynccnt)
#    define HAVE_WAITASYNC_BUILTIN 1
#  endif
#endif

// one 16-byte global -> LDS async copy per lane (ASYNCcnt-tracked)
__device__ __forceinline__ void stage_async16(const float* g, float* l) {
#ifdef HAVE_ASYNC_BUILTIN
    __builtin_amdgcn_global_load_async_to_lds_b128(
        (gv4i_p)(uintptr_t)g,
        (lv4i_p)l,
        /*offset=*/0, /*cpol=*/0);
#else
    unsigned lofs = (unsigned)(uintptr_t)(__attribute__((address_space(3))) float*)l;
    unsigned long long ga = (unsigned long long)(uintptr_t)g;
    asm volatile("global_load_async_to_lds_b128 %0, %1, off"
                 :: "v"(lofs), "v"(ga) : "memory");
#endif
}

__device__ __forceinline__ void wait_async0() {
#ifdef HAVE_WAITASYNC_BUILTIN
    __builtin_amdgcn_s_wait_asynccnt(0);
#else
    asm volatile("s_wait_asynccnt 0x0" ::: "memory");
#endif
}

__global__ void build_idct_matrix(float* __restrict__ Mmat) {
    const int c = blockIdx.x;    // output channel (row of M)
    const int k = threadIdx.x;   // input channel  (col of M)
    const float PI = 3.14159265358979323846f;
    float val;
    if (k == 0) {
        val = 1.0f;
    } else {
        val = 2.0f * cosf(PI * (float)k * ((float)c + 0.5f) * (1.0f / (float)C_DIM));
    }
    Mmat[c * C_DIM + k] = val;
}

__global__ __launch_bounds__(256) void idct_gemm(const float* __restrict__ ip,
                                                 const float* __restrict__ Mmat,
                                                 float* __restrict__ out) {
    // double-buffered B tile: [buf][k-row within chunk][pixel], padded rows
    __shared__ __align__(16) float Bs[2][KC][LDSS];

    const int tid  = threadIdx.x;
    const int lane = tid & 31;
    const int wave = tid >> 5;          // 0..7 -> owns channels [32*wave, 32*wave+32)
    const int hl   = lane >> 4;         // half-wave select (0/1)
    const int ln   = lane & 15;

    const int g0 = blockIdx.x * NT;     // global pixel tile base
    const int b  = g0 >> 14;            // 16384 pixels per image; tiles never straddle
    const int p0 = g0 & (HW - 1);

    const float* __restrict__ src = ip  + (size_t)b * C_DIM * HW + p0;
    float*       __restrict__ dst = out + (size_t)b * C_DIM * HW + p0;

    // B staging assignment: 256 threads x 16B = 32x32 tile
    const int lrow = tid >> 3;          // 0..31 (channel within chunk)
    const int lcol = (tid & 7) * 4;     // 0,4,...,28 (pixel)

    v8f acc[2][2] = {};                 // [m-tile][n-tile] accumulators

    // prime buffer 0 with K-chunk 0 (async global -> LDS)
    stage_async16(src + (size_t)lrow * HW + lcol, &Bs[0][lrow][lcol]);
    wait_async0();
    __syncthreads();

    for (int kc = 0; kc < C_DIM / KC; ++kc) {
        const int cur = kc & 1;

        // prefetch next K-chunk into the other buffer (uniform branch)
        if (kc + 1 < C_DIM / KC) {
            stage_async16(src + ((size_t)(kc + 1) * KC + lrow) * HW + lcol,
                          &Bs[cur ^ 1][lrow][lcol]);
        }

        // A fragments for this chunk (ISA 16x4 f32 A layout:
        // lane<16 holds K=k0,k0+1; lane>=16 holds K=k0+2,k0+3 of row m=ln)
        v2f afr[2][8];
        #pragma unroll
        for (int mt = 0; mt < 2; ++mt) {
            const int m = (wave * 2 + mt) * 16 + ln;
            const float* arow = Mmat + (size_t)m * C_DIM + kc * KC + 2 * hl;
            #pragma unroll
            for (int ks = 0; ks < 8; ++ks) {
                afr[mt][ks] = *reinterpret_cast<const v2f*>(arow + ks * 4);
            }
        }

        #pragma unroll
        for (int ks = 0; ks < 8; ++ks) {
            const int kr = ks * 4 + 2 * hl;   // B layout mirrors A's K split
            #pragma unroll
            for (int nt = 0; nt < 2; ++nt) {
                v2f bf;
                bf.x = Bs[cur][kr    ][nt * 16 + ln];
                bf.y = Bs[cur][kr + 1][nt * 16 + ln];
                #pragma unroll
                for (int mt = 0; mt < 2; ++mt) {
                    acc[mt][nt] = __builtin_amdgcn_wmma_f32_16x16x4_f32(
                        /*neg_a=*/false, afr[mt][ks],
                        /*neg_b=*/false, bf,
                        /*c_mod=*/(short)0, acc[mt][nt],
                        /*reuse_a=*/false, /*reuse_b=*/false);
                }
            }
        }
        // drain this wave's async stores to LDS, then block-wide barrier:
        // after the barrier every wave's staged data is visible
        wait_async0();
        __syncthreads();
    }

    // writeback: C/D layout -> VGPR r holds row (base + r) for lanes 0-15,
    // row (base + 8 + r) for lanes 16-31; N = lane%16 within n-tile
    #pragma unroll
    for (int mt = 0; mt < 2; ++mt) {
        const int cbase = (wave * 2 + mt) * 16 + hl * 8;
        #pragma unroll
        for (int r = 0; r < 8; ++r) {
            #pragma unroll
            for (int nt = 0; nt < 2; ++nt) {
                dst[(size_t)(cbase + r) * HW + nt * 16 + ln] = acc[mt][nt][r];
            }
        }
    }
}

extern "C" void kernel_launch(void* const* d_in, const int* in_sizes, int n_in,
                              void* d_out, int out_size, void* d_ws, size_t ws_size,
                              hipStream_t stream) {
    const float* ip  = (const float*)d_in[0];
    float*       out = (float*)d_out;
    float*       Mm  = (float*)d_ws;          // 256*256*4 = 256 KB scratch

    // build the DCT-III matrix (deterministic, recomputed every call)
    build_idct_matrix<<<C_DIM, C_DIM, 0, stream>>>(Mm);

    const int nPix = in_sizes[0] / C_DIM;     // b*H*W = 131072
    const int nBlocks = nPix / NT;            // 4096
    idct_gemm<<<nBlocks, 256, 0, stream>>>(ip, Mm, out);
}